// Kmeans_20014547599909
// MI455X (gfx1250) — compile-verified
//
#include <hip/hip_runtime.h>
#include <stdint.h>

#define B_   256
#define W_   14
#define HW_  196
#define C_   512
#define ITERS 10   // centroid updates; ITERS+1 assignment steps

typedef float v4f __attribute__((ext_vector_type(4)));  // native vector for NT b128 stores

__device__ __forceinline__ float sqdist(float2 p, float2 c) {
  float dx = p.x - c.x;
  float dy = p.y - c.y;
  // match JAX: (dx*dx) + (dy*dy), no FMA contraction
  return __fadd_rn(__fmul_rn(dx, dx), __fmul_rn(dy, dy));
}

// Kernel 1: one block per batch image.
// Phase A: per-channel argmax over 196 pixels. Each thread owns channels
//          2t and 2t+1 via one float2 (b64) load per pixel -> coalesced
//          256B/wave/pixel, half the load instructions of b32.
// Phase B: 2-means on the 512 (row,col) points in LDS. All partial sums are
//          small exact integers in fp32 -> reduction order irrelevant.
__global__ void __launch_bounds__(256)
argmax_kmeans_kernel(const float* __restrict__ in, unsigned char* __restrict__ assignOut) {
  const int b   = blockIdx.x;
  const int tid = threadIdx.x;           // 256 threads, 8 wave32

  __shared__ float2 pts[C_];
  __shared__ float  red[6];              // s0x s0y n0 s1x s1y n1
  __shared__ float2 cent[2];

  const float2* base2 = (const float2*)(in + (size_t)b * HW_ * C_) + tid; // ch 2t,2t+1

  float2 v0 = base2[0];
  float best0 = v0.x, best1 = v0.y;
  int   bi0 = 0, bi1 = 0;
  for (int p = 1; p < HW_; ++p) {
    float2 v = base2[(size_t)p * (C_ / 2)];
    if (v.x > best0) { best0 = v.x; bi0 = p; }   // strict > : first-max (jnp.argmax)
    if (v.y > best1) { best1 = v.y; bi1 = p; }
  }
  pts[2 * tid]     = make_float2((float)(bi0 / W_), (float)(bi0 % W_));
  pts[2 * tid + 1] = make_float2((float)(bi1 / W_), (float)(bi1 % W_));
  __syncthreads();

  float2 c0 = pts[0], c1 = pts[1];       // deterministic init
  const float2 p0 = pts[2 * tid], p1 = pts[2 * tid + 1];
  int a0 = 0, a1 = 0;

  for (int it = 0; it <= ITERS; ++it) {
    a0 = (sqdist(p0, c1) < sqdist(p0, c0)) ? 1 : 0;  // tie -> cluster 0 (argmin)
    a1 = (sqdist(p1, c1) < sqdist(p1, c0)) ? 1 : 0;
    if (it == ITERS) break;                          // last step: assignment only

    if (tid < 6) red[tid] = 0.0f;
    __syncthreads();

    float s[6] = {0.f, 0.f, 0.f, 0.f, 0.f, 0.f};
    if (a0) { s[3] += p0.x; s[4] += p0.y; s[5] += 1.f; }
    else    { s[0] += p0.x; s[1] += p0.y; s[2] += 1.f; }
    if (a1) { s[3] += p1.x; s[4] += p1.y; s[5] += 1.f; }
    else    { s[0] += p1.x; s[1] += p1.y; s[2] += 1.f; }

    #pragma unroll
    for (int i = 0; i < 6; ++i) {        // wave32 tree-reduce, then LDS f32 atomic
      float v = s[i];
      #pragma unroll
      for (int off = 16; off > 0; off >>= 1) v += __shfl_down(v, off, 32);
      if ((tid & 31) == 0) atomicAdd(&red[i], v);    // ds_add_f32
    }
    __syncthreads();

    if (tid == 0) {
      float n0 = red[2], n1 = red[5];
      float2 nc0 = c0, nc1 = c1;
      if (n0 > 0.f) { nc0.x = red[0] / n0; nc0.y = red[1] / n0; } // exact int/int
      if (n1 > 0.f) { nc1.x = red[3] / n1; nc1.y = red[4] / n1; }
      cent[0] = nc0; cent[1] = nc1;
    }
    __syncthreads();
    c0 = cent[0]; c1 = cent[1];
  }

  assignOut[(size_t)b * C_ + 2 * tid]     = (unsigned char)a0;
  assignOut[(size_t)b * C_ + 2 * tid + 1] = (unsigned char)a1;
}

// Kernel 2: NHWC -> NCHW masked transpose through a 32x32 LDS tile.
// Load side: CDNA5 async global->LDS copies (ASYNCcnt path), pitch-33 rows.
// Store side: each thread emits one v4f (b128) NT store per output half
// (lane = cRow*8 + hwChunk -> 4 contiguous 128B segments per wave), so the
// 205 MB output stream uses b128 stores and doesn't evict the L2-resident
// input (102 MB < 192 MB global L2).
__global__ void __launch_bounds__(256)
masked_transpose_kernel(const float* __restrict__ in,
                        const unsigned char* __restrict__ assign,
                        float* __restrict__ out) {
  __shared__ float tile[32][33];
  const int b      = blockIdx.z;
  const int cBase  = blockIdx.y * 32;
  const int hwBase = blockIdx.x * 32;
  const int tx = threadIdx.x;            // 0..31
  const int ty = threadIdx.y;            // 0..7

  const float*   src     = in + (size_t)b * HW_ * C_;
  const unsigned ldsBase = (unsigned)(uintptr_t)&tile[0][0];

  #pragma unroll
  for (int j = 0; j < 4; ++j) {
    const int hwl = ty + j * 8;
    const int hw  = hwBase + hwl;
    if (hw < HW_) {
      unsigned lds = ldsBase + (unsigned)((hwl * 33 + tx) * 4);
      uint64_t g   = (uint64_t)(uintptr_t)(src + (size_t)hw * C_ + cBase + tx);
      asm volatile("global_load_async_to_lds_b32 %0, %1, off"
                   :: "v"(lds), "v"(g) : "memory");
    }
  }
  asm volatile("s_wait_asynccnt 0" ::: "memory");  // our wave's LDS writes done
  __syncthreads();                                  // all waves' tiles visible

  const size_t half = (size_t)B_ * C_ * HW_;
  const int tid     = ty * 32 + tx;
  const int hwChunk = tid & 7;           // 8 float4 chunks cover 32 hw
  const int cIdx    = tid >> 3;          // 32 channels
  const int hw4     = hwBase + hwChunk * 4;

  if (hw4 < HW_) {                       // HW_ % 4 == 0 -> full float4 in range
    v4f q;
    q.x = tile[hwChunk * 4 + 0][cIdx];   // conflict-free: banks hwChunk*4+cIdx
    q.y = tile[hwChunk * 4 + 1][cIdx];
    q.z = tile[hwChunk * 4 + 2][cIdx];
    q.w = tile[hwChunk * 4 + 3][cIdx];

    const int c = cBase + cIdx;
    const unsigned char a = assign[(size_t)b * C_ + c];
    const size_t o = ((size_t)b * C_ + c) * (size_t)HW_ + hw4;  // 16B aligned

    const v4f zero = (v4f)(0.f);
    v4f q0 = a ? zero : q;
    v4f q1 = a ? q : zero;
    __builtin_nontemporal_store(q0, (v4f*)(out + o));
    __builtin_nontemporal_store(q1, (v4f*)(out + half + o));
  }
}

extern "C" void kernel_launch(void* const* d_in, const int* in_sizes, int n_in,
                              void* d_out, int out_size, void* d_ws, size_t ws_size,
                              hipStream_t stream) {
  const float* in = (const float*)d_in[0];
  float* out = (float*)d_out;
  unsigned char* assign = (unsigned char*)d_ws;   // 256*512 = 128 KB scratch

  argmax_kmeans_kernel<<<B_, 256, 0, stream>>>(in, assign);

  dim3 grid((HW_ + 31) / 32, C_ / 32, B_);        // (7, 16, 256)
  dim3 block(32, 8);
  masked_transpose_kernel<<<grid, block, 0, stream>>>(in, assign, out);
}